// _VQC2Model_8160437862656
// MI455X (gfx1250) — compile-verified
//
#include <hip/hip_runtime.h>
#include <math.h>

typedef float v2f __attribute__((ext_vector_type(2)));
typedef float v8f __attribute__((ext_vector_type(8)));

#define WAVES_PER_BLOCK 4
#define PIf 3.14159265358979323846f

// ---------------------------------------------------------------------------
// Setup: build the fixed 32x32 quadratic-form matrix M from params.
//   A = W * D_cz  (W = 2 layers of RY(params)+CNOT ring, D_cz = CZ signs)
//   M[p][q] = sum_r z0[r] * A[r][p] * A[r][q]
// One wave: lane k holds state amplitude index k (DIM == warpSize == 32).
// ---------------------------------------------------------------------------
__global__ void vqc_setup_kernel(const float* __restrict__ params,
                                 float* __restrict__ Mout) {
    __shared__ float A[32][33];
    const int lane = threadIdx.x;  // 0..31

    // gate cos/sin: g = l*5 + i, theta = params[l][i][0]
    float gc[10], gs[10];
#pragma unroll
    for (int g = 0; g < 10; ++g) {
        float th = params[g * 2];
        gs[g] = __sinf(0.5f * th);
        gc[g] = __cosf(0.5f * th);
    }

    for (int j = 0; j < 32; ++j) {
        float val = (lane == j) ? 1.0f : 0.0f;
#pragma unroll
        for (int l = 0; l < 2; ++l) {
#pragma unroll
            for (int i = 0; i < 5; ++i) {  // RY(params[l][i][0]) on wire i
                const int g = l * 5 + i;
                const int mask = 1 << (4 - i);
                float ov = __shfl(val, lane ^ mask, 32);
                int bit = (lane >> (4 - i)) & 1;
                // bit==0: c*val - s*ov ; bit==1: c*val + s*ov
                val = bit ? fmaf(gs[g], ov, gc[g] * val)
                          : fmaf(-gs[g], ov, gc[g] * val);
            }
#pragma unroll
            for (int i = 0; i < 5; ++i) {  // CNOT(control=i, target=(i+1)%5)
                const int t = (i + 1) % 5;
                int cbit = (lane >> (4 - i)) & 1;
                int src = cbit ? (lane ^ (1 << (4 - t))) : lane;
                val = __shfl(val, src, 32);
            }
        }
        // CZ ring sign for column j
        float dcz = 1.0f;
#pragma unroll
        for (int i = 0; i < 5; ++i) {
            int b1 = (j >> (4 - i)) & 1;
            int b2 = (j >> (4 - (i + 1) % 5)) & 1;
            if (b1 & b2) dcz = -dcz;
        }
        A[lane][j] = dcz * val;
    }
    __syncthreads();

    // M[p][q], q = lane
    for (int p = 0; p < 32; ++p) {
        float m = 0.0f;
#pragma unroll
        for (int r = 0; r < 32; ++r) {
            float zr = (r & 16) ? -1.0f : 1.0f;  // Z on wire 0 (MSB)
            m = fmaf(zr * A[r][p], A[r][lane], m);
        }
        Mout[p * 32 + lane] = m;
    }
}

// ---------------------------------------------------------------------------
// Main kernel: one wave handles 16 batch elements.
//   S (16x32, re & im) built as product-state amplitudes, directly in the
//   f32 WMMA A-operand layout (lane half owns K offset +2).
//   T = S @ M via 8 k-steps x 2 n-tiles of V_WMMA_F32_16X16X4_F32.
//   out_b = sum_n T[b][n] * S[b][n]  (re) + same (im), reduced via LDS+shfl.
// ---------------------------------------------------------------------------
__global__ __launch_bounds__(WAVES_PER_BLOCK * 32) void vqc_main_kernel(
    const float* __restrict__ x, const float* __restrict__ Mq,
    float* __restrict__ out, int batch) {
    const int lane = threadIdx.x & 31;
    const int wave = threadIdx.x >> 5;
    const int bl = lane & 15;  // batch row within group
    const int hi = lane >> 4;  // half-wave: K/N offset of +2 per ISA layout
    const int group = blockIdx.x * WAVES_PER_BLOCK + wave;
    const int e = group * 16 + bl;
    const int el = (e < batch) ? e : (batch - 1);  // clamp loads, keep EXEC full

    // ---- per-wire c,s ----
    float cc[5], ss[5];
#pragma unroll
    for (int w = 0; w < 5; ++w) {
        float th = 0.5f * PIf * x[el * 5 + w];
        __sincosf(th, &ss[w], &cc[w]);
    }
    // qubit amplitudes: q_w[0] = (c^2, -c*s), q_w[1] = (c*s, s^2)
    float q0r[2] = {cc[0] * cc[0], cc[0] * ss[0]};
    float q0i[2] = {-cc[0] * ss[0], ss[0] * ss[0]};
    float q1r[2] = {cc[1] * cc[1], cc[1] * ss[1]};
    float q1i[2] = {-cc[1] * ss[1], ss[1] * ss[1]};
    float q2r[2] = {cc[2] * cc[2], cc[2] * ss[2]};
    float q2i[2] = {-cc[2] * ss[2], ss[2] * ss[2]};
    // tail factor t_j = q3[hi] * q4[j]   (b3 = hi, b4 = j)
    float q3r = hi ? cc[3] * ss[3] : cc[3] * cc[3];
    float q3i = hi ? ss[3] * ss[3] : -cc[3] * ss[3];
    float q4r[2] = {cc[4] * cc[4], cc[4] * ss[4]};
    float q4i[2] = {-cc[4] * ss[4], ss[4] * ss[4]};
    float tr[2], ti[2];
#pragma unroll
    for (int j = 0; j < 2; ++j) {
        tr[j] = q3r * q4r[j] - q3i * q4i[j];
        ti[j] = q3r * q4i[j] + q3i * q4r[j];
    }
    // prefix products p_kk = q0[b0]*q1[b1]*q2[b2], kk = b0*4+b1*2+b2
    float p01r[4], p01i[4];
#pragma unroll
    for (int b0 = 0; b0 < 2; ++b0)
#pragma unroll
        for (int b1 = 0; b1 < 2; ++b1) {
            p01r[b0 * 2 + b1] = q0r[b0] * q1r[b1] - q0i[b0] * q1i[b1];
            p01i[b0 * 2 + b1] = q0r[b0] * q1i[b1] + q0i[b0] * q1r[b1];
        }
    // A-operand values: are[kk] = (S_re[bl][4kk+2hi], S_re[bl][4kk+2hi+1])
    v2f are[8], aim[8];
#pragma unroll
    for (int kk = 0; kk < 8; ++kk) {
        const int b01 = kk >> 1, b2 = kk & 1;
        float pr = p01r[b01] * q2r[b2] - p01i[b01] * q2i[b2];
        float pi = p01r[b01] * q2i[b2] + p01i[b01] * q2r[b2];
#pragma unroll
        for (int j = 0; j < 2; ++j) {
            are[kk][j] = pr * tr[j] - pi * ti[j];
            aim[kk][j] = pr * ti[j] + pi * tr[j];
        }
    }

    // ---- T = S @ M via f32 WMMA, 8 K-steps x 2 N-tiles x {re,im} ----
    v8f accRe0 = {}, accRe1 = {}, accIm0 = {}, accIm1 = {};
#pragma unroll
    for (int kk = 0; kk < 8; ++kk) {
        const int krow = 4 * kk + 2 * hi;  // B layout: halves hold K, K+2
        v2f b0v, b1v;
        b0v[0] = Mq[krow * 32 + bl];
        b0v[1] = Mq[(krow + 1) * 32 + bl];
        b1v[0] = Mq[krow * 32 + 16 + bl];
        b1v[1] = Mq[(krow + 1) * 32 + 16 + bl];
        accRe0 = __builtin_amdgcn_wmma_f32_16x16x4_f32(false, are[kk], false,
                                                       b0v, (short)0, accRe0,
                                                       false, false);
        accRe1 = __builtin_amdgcn_wmma_f32_16x16x4_f32(false, are[kk], false,
                                                       b1v, (short)0, accRe1,
                                                       false, false);
        accIm0 = __builtin_amdgcn_wmma_f32_16x16x4_f32(false, aim[kk], false,
                                                       b0v, (short)0, accIm0,
                                                       false, false);
        accIm1 = __builtin_amdgcn_wmma_f32_16x16x4_f32(false, aim[kk], false,
                                                       b1v, (short)0, accIm1,
                                                       false, false);
    }

    // ---- stage T through LDS (D layout -> [b][n]) and contract with S ----
    __shared__ float T[WAVES_PER_BLOCK][2][16][33];
#pragma unroll
    for (int r = 0; r < 8; ++r) {
        const int m = r + 8 * hi;  // C/D layout: VGPR r holds rows r, r+8
        T[wave][0][m][bl] = accRe0[r];
        T[wave][0][m][16 + bl] = accRe1[r];
        T[wave][1][m][bl] = accIm0[r];
        T[wave][1][m][16 + bl] = accIm1[r];
    }
    __syncthreads();

    float sum = 0.0f;
#pragma unroll
    for (int kk = 0; kk < 8; ++kk) {
        const int n = 4 * kk + 2 * hi;  // dims this lane owns in A layout
        sum = fmaf(T[wave][0][bl][n], are[kk][0], sum);
        sum = fmaf(T[wave][0][bl][n + 1], are[kk][1], sum);
        sum = fmaf(T[wave][1][bl][n], aim[kk][0], sum);
        sum = fmaf(T[wave][1][bl][n + 1], aim[kk][1], sum);
    }
    sum += __shfl(sum, lane ^ 16, 32);  // combine the two half-wave partials
    if (hi == 0 && e < batch) out[e] = sum;
}

// ---------------------------------------------------------------------------
extern "C" void kernel_launch(void* const* d_in, const int* in_sizes, int n_in,
                              void* d_out, int out_size, void* d_ws,
                              size_t ws_size, hipStream_t stream) {
    const float* x = (const float*)d_in[0];       // (BATCH, 5) f32
    const float* params = (const float*)d_in[1];  // (2, 5, 2) f32
    float* out = (float*)d_out;                   // (BATCH,) f32
    float* Mq = (float*)d_ws;                     // 32*32 f32 scratch

    const int batch = in_sizes[0] / 5;

    vqc_setup_kernel<<<1, 32, 0, stream>>>(params, Mq);

    const int groups = (batch + 15) / 16;
    const int blocks = (groups + WAVES_PER_BLOCK - 1) / WAVES_PER_BLOCK;
    vqc_main_kernel<<<blocks, WAVES_PER_BLOCK * 32, 0, stream>>>(x, Mq, out,
                                                                 batch);
}